// TransitionUp_16716012716554
// MI455X (gfx1250) — compile-verified
//
#include <hip/hip_runtime.h>

typedef __attribute__((ext_vector_type(2))) float v2f;
typedef __attribute__((ext_vector_type(8))) float v8f;

#define B_    4
#define N1_   4096
#define N2_   16384
#define CIN_  512
#define COUT_ 256
#define BN_EPS_ 1e-5f

// ---------------------------------------------------------------------------
// gfx1250 async global->LDS copy (ASYNCcnt-tracked); per-lane scatter.
// LDS byte offset = low 32 bits of the generic pointer (flat aperture rule).
// ---------------------------------------------------------------------------
__device__ __forceinline__ void async_g2l_b32(const float* g, const float* l) {
  asm volatile("global_load_async_to_lds_b32 %0, %1, off"
               :: "v"((uint32_t)(size_t)l), "v"((uint64_t)(size_t)g)
               : "memory");
}
__device__ __forceinline__ void wait_async0() {
  asm volatile("s_wait_asynccnt 0x0" ::: "memory");
}

// ---------------------------------------------------------------------------
// GEMM with bias: Y[M,N] = A[M,K] @ W[K,N] + bias[N]
// Block = 8 waves; block computes a 128(M) x 16(N) tile of Y.
// The K x 16 strip of W is staged into LDS once (async DMA), shared by all
// 8 waves; each wave then runs the K/4 chain of V_WMMA_F32_16X16X4_F32.
// A frag (16x4 f32): lane l -> row (l&15), VGPR v -> K = (l>>4)*2 + v
// B frag (4x16 f32): lane l -> col (l&15), VGPR v -> K = (l>>4)*2 + v
// C/D (16x16 f32):   VGPR r -> row r (lanes 0-15) / row r+8 (lanes 16-31)
// ---------------------------------------------------------------------------
template <int K>
__global__ __launch_bounds__(256) void gemm_bias_wmma(
    const float* __restrict__ A, const float* __restrict__ W,
    const float* __restrict__ bias, float* __restrict__ Y,
    int M, int N)
{
  __shared__ float wlds[K * 16];               // <= 32 KB strip of W

  const int ntiles = N >> 4;
  const int nt   = blockIdx.x % ntiles;        // N-tile index
  const int mblk = blockIdx.x / ntiles;        // 128-row super-tile index
  const int tid  = threadIdx.x;

  // ---- stage W[:, nt*16 .. nt*16+15] into LDS via async DMA ----
  {
    const float* __restrict__ src = W + nt * 16;
#pragma unroll 4
    for (int i = tid; i < K * 16; i += 256) {
      const int kk = i >> 4, cc = i & 15;
      async_g2l_b32(src + (size_t)kk * N + cc, &wlds[i]);
    }
    wait_async0();
  }
  __syncthreads();

  const int wv   = tid >> 5;
  const int lane = tid & 31;
  const int half = lane >> 4;                  // 0: K pair {0,1}, 1: {2,3}
  const int l15  = lane & 15;
  const int row0 = mblk * 128 + wv * 16;

  v8f acc = {};
  const float* __restrict__ Arow = A + (size_t)(row0 + l15) * K;

#pragma unroll 8
  for (int k = 0; k < K; k += 4) {
    const int ka = k + half * 2;
    if ((k & 31) == 0 && k + 128 < K)          // compile-time under unroll
      __builtin_prefetch(Arow + k + 128, 0, 1);
    v2f a, b;
    a.x = Arow[ka];
    a.y = Arow[ka + 1];
    b.x = wlds[ka * 16 + l15];
    b.y = wlds[(ka + 1) * 16 + l15];
    acc = __builtin_amdgcn_wmma_f32_16x16x4_f32(
        /*neg_a=*/false, a, /*neg_b=*/false, b,
        /*c_mod=*/(short)0, acc, /*reuse_a=*/false, /*reuse_b=*/false);
  }

  const float bv = bias[nt * 16 + l15];
#pragma unroll
  for (int r = 0; r < 8; ++r) {
    const int row = row0 + half * 8 + r;
    Y[(size_t)row * N + nt * 16 + l15] = acc[r] + bv;
  }
}

// ---------------------------------------------------------------------------
// Per-channel batch stats: mean + rsqrt(var + eps). One block per channel.
// ---------------------------------------------------------------------------
__global__ __launch_bounds__(256) void colstats(
    const float* __restrict__ Y, int M, int N,
    float* __restrict__ mean, float* __restrict__ rstd)
{
  const int c = blockIdx.x;
  float s = 0.f, s2 = 0.f;
  for (int r = threadIdx.x; r < M; r += blockDim.x) {
    const float v = Y[(size_t)r * N + c];
    s += v; s2 += v * v;
  }
  __shared__ float sh[256], sh2[256];
  sh[threadIdx.x] = s; sh2[threadIdx.x] = s2;
  __syncthreads();
  for (int off = 128; off > 0; off >>= 1) {
    if ((int)threadIdx.x < off) {
      sh[threadIdx.x]  += sh[threadIdx.x + off];
      sh2[threadIdx.x] += sh2[threadIdx.x + off];
    }
    __syncthreads();
  }
  if (threadIdx.x == 0) {
    const float m   = sh[0] / (float)M;
    const float var = sh2[0] / (float)M - m * m;
    mean[c] = m;
    rstd[c] = rsqrtf(var + BN_EPS_);
  }
}

// ---------------------------------------------------------------------------
// Brute-force 3-NN: grid.y = batch, one thread per query point.
// p1[b] (4096 x 3, 48 KB SoA) async-staged in LDS; branchless top-3 insert.
// ---------------------------------------------------------------------------
__global__ __launch_bounds__(256) void knn3(
    const float* __restrict__ p1, const float* __restrict__ p2,
    int* __restrict__ idx, float* __restrict__ wgt)
{
  __shared__ float sx[N1_], sy[N1_], sz[N1_];
  const int b = blockIdx.y;
  const float* __restrict__ P1 = p1 + (size_t)b * N1_ * 3;
  for (int i = threadIdx.x; i < N1_; i += blockDim.x) {
    async_g2l_b32(P1 + i * 3 + 0, &sx[i]);
    async_g2l_b32(P1 + i * 3 + 1, &sy[i]);
    async_g2l_b32(P1 + i * 3 + 2, &sz[i]);
  }
  wait_async0();
  __syncthreads();

  const int q = blockIdx.x * blockDim.x + threadIdx.x;
  const float* __restrict__ P2 = p2 + ((size_t)b * N2_ + q) * 3;
  const float qx = P2[0], qy = P2[1], qz = P2[2];

  float d0 = 3.0e38f, d1 = 3.0e38f, d2 = 3.0e38f;
  int   i0 = 0,       i1 = 0,       i2 = 0;

  for (int i = 0; i < N1_; ++i) {
    const float dx = qx - sx[i];
    const float dy = qy - sy[i];
    const float dz = qz - sz[i];
    const float d  = dx * dx + dy * dy + dz * dz;
    const bool c0 = d < d0;
    const bool c1 = d < d1;
    const bool c2 = d < d2;
    const float nd2 = c1 ? d1 : (c2 ? d : d2);
    const int   ni2 = c1 ? i1 : (c2 ? i : i2);
    const float nd1 = c0 ? d0 : (c1 ? d : d1);
    const int   ni1 = c0 ? i0 : (c1 ? i : i1);
    const float nd0 = c0 ? d  : d0;
    const int   ni0 = c0 ? i  : i0;
    d2 = nd2; i2 = ni2; d1 = nd1; i1 = ni1; d0 = nd0; i0 = ni0;
  }

  const float r0 = 1.f / (sqrtf(fmaxf(d0, 0.f)) + 1e-8f);
  const float r1 = 1.f / (sqrtf(fmaxf(d1, 0.f)) + 1e-8f);
  const float r2 = 1.f / (sqrtf(fmaxf(d2, 0.f)) + 1e-8f);
  const float inv = 1.f / (r0 + r1 + r2);

  const size_t o = ((size_t)b * N2_ + q) * 3;
  idx[o + 0] = i0; idx[o + 1] = i1; idx[o + 2] = i2;
  wgt[o + 0] = r0 * inv; wgt[o + 1] = r1 * inv; wgt[o + 2] = r2 * inv;
}

// ---------------------------------------------------------------------------
// out = sum_k w_k * relu(bn1(y1[idx_k])) + relu(bn2(y2)); one thread/element.
// ---------------------------------------------------------------------------
__global__ __launch_bounds__(256) void finalize(
    const float* __restrict__ y1,
    const float* __restrict__ mean1, const float* __restrict__ rstd1,
    const float* __restrict__ g1, const float* __restrict__ be1,
    const float* __restrict__ y2,
    const float* __restrict__ mean2, const float* __restrict__ rstd2,
    const float* __restrict__ g2, const float* __restrict__ be2,
    const int* __restrict__ idx, const float* __restrict__ wgt,
    float* __restrict__ out)
{
  const size_t t = (size_t)blockIdx.x * blockDim.x + threadIdx.x; // [0, B*N2*C)
  const int    c   = (int)(t & (COUT_ - 1));
  const size_t row = t >> 8;           // COUT_ == 256
  const int    b   = (int)(row >> 14); // N2_  == 16384

  const float m1 = mean1[c], rs1 = rstd1[c], gg1 = g1[c], bb1 = be1[c];
  const size_t o3 = row * 3;

  float acc = 0.f;
#pragma unroll
  for (int k = 0; k < 3; ++k) {
    const int   id = idx[o3 + k];
    const float w  = wgt[o3 + k];
    float v = y1[((size_t)b * N1_ + id) * COUT_ + c];
    v = fmaxf((v - m1) * rs1 * gg1 + bb1, 0.f);
    acc += w * v;
  }

  float h = y2[t];
  h = fmaxf((h - mean2[c]) * rstd2[c] * g2[c] + be2[c], 0.f);
  out[t] = acc + h;
}

// ---------------------------------------------------------------------------
extern "C" void kernel_launch(void* const* d_in, const int* in_sizes, int n_in,
                              void* d_out, int out_size, void* d_ws, size_t ws_size,
                              hipStream_t stream) {
  const float* x1    = (const float*)d_in[0];
  const float* p1    = (const float*)d_in[1];
  const float* x2    = (const float*)d_in[2];
  const float* p2    = (const float*)d_in[3];
  const float* W_in  = (const float*)d_in[4];
  const float* b_in  = (const float*)d_in[5];
  const float* g_in  = (const float*)d_in[6];
  const float* be_in = (const float*)d_in[7];
  const float* W_out = (const float*)d_in[8];
  const float* b_out = (const float*)d_in[9];
  const float* g_out = (const float*)d_in[10];
  const float* be_out= (const float*)d_in[11];
  float* out = (float*)d_out;

  // workspace layout (floats)
  float* y1    = (float*)d_ws;                      // B*N1*COUT
  float* y2    = y1 + (size_t)B_ * N1_ * COUT_;     // B*N2*COUT
  float* mean1 = y2 + (size_t)B_ * N2_ * COUT_;
  float* rstd1 = mean1 + COUT_;
  float* mean2 = rstd1 + COUT_;
  float* rstd2 = mean2 + COUT_;
  float* wgt   = rstd2 + COUT_;                     // B*N2*3
  int*   idx   = (int*)(wgt + (size_t)B_ * N2_ * 3);

  // GEMM1: [B*N1, CIN] x [CIN, COUT]
  {
    const int M = B_ * N1_, N = COUT_;
    gemm_bias_wmma<CIN_><<<(M / 128) * (N / 16), 256, 0, stream>>>(
        x1, W_in, b_in, y1, M, N);
    colstats<<<N, 256, 0, stream>>>(y1, M, N, mean1, rstd1);
  }

  // 3-NN + interpolation weights
  knn3<<<dim3(N2_ / 256, B_), 256, 0, stream>>>(p1, p2, idx, wgt);

  // GEMM2: [B*N2, COUT] x [COUT, COUT]
  {
    const int M = B_ * N2_, N = COUT_;
    gemm_bias_wmma<COUT_><<<(M / 128) * (N / 16), 256, 0, stream>>>(
        x2, W_out, b_out, y2, M, N);
    colstats<<<N, 256, 0, stream>>>(y2, M, N, mean2, rstd2);
  }

  // fused interp + BN/ReLU + add
  {
    const size_t total = (size_t)B_ * N2_ * COUT_;
    finalize<<<(unsigned)(total / 256), 256, 0, stream>>>(
        y1, mean1, rstd1, g_in, be_in,
        y2, mean2, rstd2, g_out, be_out,
        idx, wgt, out);
  }

  // tuple second element: p2 passthrough
  hipMemcpyAsync(out + (size_t)B_ * N2_ * COUT_, p2,
                 (size_t)B_ * N2_ * 3 * sizeof(float),
                 hipMemcpyDeviceToDevice, stream);
}